// QNet_4776003633689
// MI455X (gfx1250) — compile-verified
//
#include <hip/hip_runtime.h>

typedef float v2f __attribute__((ext_vector_type(2)));
typedef float v8f __attribute__((ext_vector_type(8)));

#if __has_builtin(__builtin_amdgcn_wmma_f32_16x16x4_f32)
#define USE_WMMA_F32 1
#else
#define USE_WMMA_F32 0
#endif

// ------------------------------------------------------------------ utilities

__global__ void k_fill(float* __restrict__ p, float v, long long n) {
    long long i = (long long)blockIdx.x * blockDim.x + threadIdx.x;
    long long st = (long long)gridDim.x * blockDim.x;
    for (; i < n; i += st) p[i] = v;
}

__device__ __forceinline__ void atomicMaxFloat(float* addr, float val) {
    unsigned int* ai = (unsigned int*)addr;
    unsigned int old = *ai;
    while (true) {
        float cur = __uint_as_float(old);
        if (cur >= val) break;
        unsigned int assumed = old;
        old = atomicCAS(ai, assumed, __float_as_uint(val));
        if (old == assumed) break;
    }
}

// --------------------------------------------------------------- WMMA GEMM
// C[M x 128] = act(A[M x K] @ B[K x 128] (+bias)); K % 4 == 0.
// Block = 8 waves; each wave owns one 16-col tile and FOUR 16-row tiles
// (64 rows), sharing the B fragment across the 4 v_wmma issues per K-step.
// Out-of-range A rows are clamped to a valid address (their outputs are
// masked at the store), so all fragment loads are unconditional b64/b32.
__global__ void __launch_bounds__(256)
k_gemm128_wmma(const float* __restrict__ A, const float* __restrict__ B,
               const float* __restrict__ bias, float* __restrict__ C,
               int M, int K, int relu)
{
    const int N = 128;
    const int TM = 4;
    int lane  = threadIdx.x;                  // 0..31
    int n0    = threadIdx.y << 4;             // column tile base (8 tiles)
    int mbase = blockIdx.x * (TM * 16);       // 64-row super-tile base
    int ncol  = n0 + (lane & 15);
    int khalf = (lane >> 4) << 1;             // 0 (lanes 0-15) or 2 (lanes 16-31)
    int msub  = lane & 15;

    size_t arow[TM];
#pragma unroll
    for (int t = 0; t < TM; ++t) {
        int mr = mbase + t * 16 + msub;
        if (mr >= M) mr = M - 1;              // clamp: garbage rows masked at store
        arow[t] = (size_t)mr * K;
    }

    v8f acc[TM];
#pragma unroll
    for (int t = 0; t < TM; ++t)
#pragma unroll
        for (int r = 0; r < 8; ++r) acc[t][r] = 0.0f;

#if USE_WMMA_F32
    for (int k = 0; k < K; k += 4) {
        v2f bv;   // B 4x16 frag: VGPR0 = K=khalf, VGPR1 = K=khalf+1
        bv.x = B[(size_t)(k + khalf)     * N + ncol];
        bv.y = B[(size_t)(k + khalf + 1) * N + ncol];
#pragma unroll
        for (int t = 0; t < TM; ++t) {
            // A 16x4 frag, two consecutive K elements -> single b64 load
            v2f av = *reinterpret_cast<const v2f*>(A + arow[t] + k + khalf);
            acc[t] = __builtin_amdgcn_wmma_f32_16x16x4_f32(false, av, false, bv,
                                                           (short)0, acc[t], false, false);
        }
    }
#else
    for (int k = 0; k < K; ++k) {
        float bvv = B[(size_t)k * N + ncol];
#pragma unroll
        for (int t = 0; t < TM; ++t) {
            int rb = mbase + t * 16 + ((lane >> 4) << 3);
#pragma unroll
            for (int r = 0; r < 8; ++r) {
                int row = rb + r;
                float avv = (row < M) ? A[(size_t)row * K + k] : 0.0f;
                acc[t][r] += avv * bvv;
            }
        }
    }
#endif
    // C/D layout: VGPR r -> row rbase+r (+8 for upper lane half), col = lane&15
#pragma unroll
    for (int t = 0; t < TM; ++t) {
        int rbase = mbase + t * 16 + ((lane >> 4) << 3);
#pragma unroll
        for (int r = 0; r < 8; ++r) {
            int row = rbase + r;
            if (row < M) {
                float v = acc[t][r];
                if (bias) v += bias[ncol];
                if (relu) v = fmaxf(v, 0.0f);
                C[(size_t)row * N + ncol] = v;
            }
        }
    }
}

// ------------------------------------------------------- GAT helper kernels

// a[n,h] = sum_c P[n, h*32+c] * att[h*32+c]; one thread per node, float4 loads
__global__ void k_head_reduce(const float* __restrict__ P, const float* __restrict__ att,
                              float* __restrict__ a, int N)
{
    __shared__ float sat[128];
    if (threadIdx.x < 128) sat[threadIdx.x] = att[threadIdx.x];
    __syncthreads();
    int n = blockIdx.x * blockDim.x + threadIdx.x;
    if (n >= N) return;
    const float* row = P + (size_t)n * 128;
#pragma unroll
    for (int h = 0; h < 4; ++h) {
        float acc = 0.0f;
#pragma unroll
        for (int c = 0; c < 32; c += 4) {
            float4 x = *reinterpret_cast<const float4*>(row + h * 32 + c);
            acc += x.x * sat[h * 32 + c] + x.y * sat[h * 32 + c + 1]
                 + x.z * sat[h * 32 + c + 2] + x.w * sat[h * 32 + c + 3];
        }
        a[(size_t)n * 4 + h] = acc;
    }
}

// V[k*4+h] = sum_c W[k*128 + h*32 + c] * att[h*32+c]   (fold att_dst into W)
__global__ void k_fold_att(const float* __restrict__ W, const float* __restrict__ att,
                           float* __restrict__ V)
{
    int idx = blockIdx.x * blockDim.x + threadIdx.x;
    if (idx >= 512) return;
    int k = idx >> 2, h = idx & 3;
    float acc = 0.0f;
#pragma unroll
    for (int c = 0; c < 32; ++c) acc += W[(size_t)k * 128 + h * 32 + c] * att[h * 32 + c];
    V[idx] = acc;
}

// a_d[n,h] = sum_k X[n,k] * V[k*4+h]; one thread per node, float4 row loads
__global__ void k_rowdot4(const float* __restrict__ X, const float* __restrict__ V,
                          float* __restrict__ out, int N)
{
    __shared__ float sv[512];
    for (int i = threadIdx.x; i < 512; i += blockDim.x) sv[i] = V[i];
    __syncthreads();
    int n = blockIdx.x * blockDim.x + threadIdx.x;
    if (n >= N) return;
    const float* row = X + (size_t)n * 128;
    float a0 = 0.f, a1 = 0.f, a2 = 0.f, a3 = 0.f;
    for (int k = 0; k < 128; k += 4) {
        float4 x = *reinterpret_cast<const float4*>(row + k);
        a0 += x.x * sv[k*4+0] + x.y * sv[(k+1)*4+0] + x.z * sv[(k+2)*4+0] + x.w * sv[(k+3)*4+0];
        a1 += x.x * sv[k*4+1] + x.y * sv[(k+1)*4+1] + x.z * sv[(k+2)*4+1] + x.w * sv[(k+3)*4+1];
        a2 += x.x * sv[k*4+2] + x.y * sv[(k+1)*4+2] + x.z * sv[(k+2)*4+2] + x.w * sv[(k+3)*4+2];
        a3 += x.x * sv[k*4+3] + x.y * sv[(k+1)*4+3] + x.z * sv[(k+2)*4+3] + x.w * sv[(k+3)*4+3];
    }
    float* o = out + (size_t)n * 4;
    o[0] = a0; o[1] = a1; o[2] = a2; o[3] = a3;
}

// pass 1: e = leaky_relu(a_s[src]+a_d[dst]); atomic max per (dst,head)
__global__ void k_edge1(const int* __restrict__ edge, int E,
                        const float* __restrict__ a_s, const float* __restrict__ a_d,
                        float* __restrict__ ebuf, float* __restrict__ mx)
{
    int idx = blockIdx.x * blockDim.x + threadIdx.x;
    if (idx >= E * 4) return;
    int e = idx >> 2, h = idx & 3;
    int s = edge[e], d = edge[E + e];
    float v = a_s[s * 4 + h] + a_d[d * 4 + h];
    v = (v >= 0.0f) ? v : 0.2f * v;
    ebuf[idx] = v;
    atomicMaxFloat(&mx[d * 4 + h], v);
}

// pass 2: ex = exp(e - mx[dst]); atomic add denominator
__global__ void k_edge2(const int* __restrict__ edge, int E,
                        float* __restrict__ ebuf, const float* __restrict__ mx,
                        float* __restrict__ den)
{
    int idx = blockIdx.x * blockDim.x + threadIdx.x;
    if (idx >= E * 4) return;
    int e = idx >> 2, h = idx & 3;
    int d = edge[E + e];
    float v = expf(ebuf[idx] - mx[d * 4 + h]);
    ebuf[idx] = v;
    atomicAdd(&den[d * 4 + h], v);
}

// pass 3: out[dst,:] += P[src,:] * softmax_weight.  32 lanes per edge,
// each lane does one b128 load of P and 4 atomic f32 adds.
__global__ void k_edge3(const int* __restrict__ edge, int E,
                        const float* __restrict__ ebuf, const float* __restrict__ den,
                        const float* __restrict__ P, float* __restrict__ out)
{
    int tid  = blockIdx.x * blockDim.x + threadIdx.x;
    int e    = tid >> 5;
    int lane = tid & 31;
    if (e >= E) return;
    int s = edge[e], d = edge[E + e];
    int c0 = lane * 4;                 // 4 consecutive channels, same head
    int h  = c0 >> 5;
    float w = ebuf[(size_t)e * 4 + h] / (den[(size_t)d * 4 + h] + 1e-16f);
    float4 pv = *reinterpret_cast<const float4*>(P + (size_t)s * 128 + c0);
    float* ob = out + (size_t)d * 128 + c0;
    atomicAdd(ob + 0, pv.x * w);
    atomicAdd(ob + 1, pv.y * w);
    atomicAdd(ob + 2, pv.z * w);
    atomicAdd(ob + 3, pv.w * w);
}

// out = residual + sum of (up to 5) relation biases, float4 granularity
__global__ void k_init_out(float4* __restrict__ out, const float4* __restrict__ res,
                           const float* b0, const float* b1, const float* b2,
                           const float* b3, const float* b4, long long n4)
{
    long long idx = (long long)blockIdx.x * blockDim.x + threadIdx.x;
    if (idx >= n4) return;
    int c = (int)(idx & 31) * 4;       // channel group within a 128-wide row
    float4 v = res[idx];
    if (b0) { v.x += b0[c]; v.y += b0[c+1]; v.z += b0[c+2]; v.w += b0[c+3]; }
    if (b1) { v.x += b1[c]; v.y += b1[c+1]; v.z += b1[c+2]; v.w += b1[c+3]; }
    if (b2) { v.x += b2[c]; v.y += b2[c+1]; v.z += b2[c+2]; v.w += b2[c+3]; }
    if (b3) { v.x += b3[c]; v.y += b3[c+1]; v.z += b3[c+2]; v.w += b3[c+3]; }
    if (b4) { v.x += b4[c]; v.y += b4[c+1]; v.z += b4[c+2]; v.w += b4[c+3]; }
    out[idx] = v;
}

// Y[n,:] = LayerNorm(relu(X[n,:])) * g + b  (one 128-thread block per row)
__global__ void k_ln_relu(const float* __restrict__ X, const float* __restrict__ g,
                          const float* __restrict__ b, float* __restrict__ Y, int N)
{
    __shared__ float red[128];
    int n = blockIdx.x, t = threadIdx.x;
    float v = fmaxf(X[(size_t)n * 128 + t], 0.0f);
    red[t] = v; __syncthreads();
    for (int s = 64; s > 0; s >>= 1) { if (t < s) red[t] += red[t + s]; __syncthreads(); }
    float m = red[0] * (1.0f / 128.0f);
    __syncthreads();
    float dv = v - m;
    red[t] = dv * dv; __syncthreads();
    for (int s = 64; s > 0; s >>= 1) { if (t < s) red[t] += red[t + s]; __syncthreads(); }
    float var = red[0] * (1.0f / 128.0f);
    Y[(size_t)n * 128 + t] = dv * rsqrtf(var + 1e-5f) * g[t] + b[t];
}

// column sums of X[N x 128] into acc[128]
__global__ void k_colsum(const float* __restrict__ X, float* __restrict__ acc, int N)
{
    int c = threadIdx.x;  // 128 threads
    float s = 0.0f;
    for (int n = blockIdx.x; n < N; n += gridDim.x) s += X[(size_t)n * 128 + c];
    atomicAdd(&acc[c], s);
}

// h_global = relu(gvec[896] @ Wg[896x128] + bg); single 128-thread block
__global__ void k_global_head(const float* __restrict__ hs, const float* __restrict__ hm,
                              const float* __restrict__ hr, const float* __restrict__ colsum,
                              float inv_n, const float* __restrict__ Wg,
                              const float* __restrict__ bg, float* __restrict__ hglob)
{
    int c = threadIdx.x;
    float acc = bg[c];
    for (int j = 0; j < 896; ++j) {
        float gv;
        if      (j < 384) gv = hs[j];              // hs[0:3] flattened
        else if (j < 640) gv = hm[j - 384];        // hm[0:2]
        else if (j < 768) gv = hr[j - 640];        // hr[0]
        else              gv = colsum[j - 768] * inv_n;  // hj.mean
        acc += gv * Wg[(size_t)j * 128 + c];
    }
    hglob[c] = fmaxf(acc, 0.0f);
}

// Q MLP: one 64-thread block per action
__global__ void k_qhead(const float* __restrict__ actions, const float* __restrict__ alpha,
                        const float* __restrict__ hj, const float* __restrict__ hglob,
                        const float* __restrict__ W1, const float* __restrict__ b1,
                        const float* __restrict__ W2, const float* __restrict__ b2,
                        const float* __restrict__ W3, const float* __restrict__ b3,
                        float* __restrict__ out, int A)
{
    __shared__ float feat[259];
    __shared__ float q1[64];
    __shared__ float q2[32];
    int a = blockIdx.x, t = threadIdx.x;
    int jid = (int)actions[a * 3 + 0];
    for (int j = t; j < 259; j += 64) {
        float v;
        if      (j < 128) v = hj[(size_t)jid * 128 + j];
        else if (j < 256) v = hglob[j - 128];
        else if (j == 256) v = actions[a * 3 + 1];
        else if (j == 257) v = actions[a * 3 + 2];
        else               v = alpha[0];
        feat[j] = v;
    }
    __syncthreads();
    float acc = b1[t];
    for (int j = 0; j < 259; ++j) acc += feat[j] * W1[(size_t)j * 64 + t];
    q1[t] = fmaxf(acc, 0.0f);
    __syncthreads();
    if (t < 32) {
        float acc2 = b2[t];
        for (int j = 0; j < 64; ++j) acc2 += q1[j] * W2[(size_t)j * 32 + t];
        q2[t] = fmaxf(acc2, 0.0f);
    }
    __syncthreads();
    if (t == 0) {
        float acc3 = b3[0];
        for (int j = 0; j < 32; ++j) acc3 += q2[j] * W3[j];
        out[a] = acc3;
    }
}

// ------------------------------------------------------------------- driver

extern "C" void kernel_launch(void* const* d_in, const int* in_sizes, int n_in,
                              void* d_out, int out_size, void* d_ws, size_t ws_size,
                              hipStream_t stream)
{
    if (n_in < 128) return;   // expects 16 data inputs + 112 param leaves
    (void)out_size; (void)ws_size;

    const float* x_job = (const float*)d_in[0];
    const float* x_sta = (const float*)d_in[1];
    const float* x_mac = (const float*)d_in[2];
    const float* x_rob = (const float*)d_in[3];
    const int* eg[10]; int nE[10];
    for (int i = 0; i < 10; ++i) { eg[i] = (const int*)d_in[4 + i]; nE[i] = in_sizes[4 + i] / 2; }
    // edge order: 0 can_load, 1 loaded, 2 will_execute, 3 execute, 4 hold,
    //             5 could_be_loaded, 6 loaded_in, 7 needs, 8 executed_by, 9 hold_by
    const float* actions = (const float*)d_in[14];
    const float* alpha   = (const float*)d_in[15];
    auto PF = [&](int i) -> const float* { return (const float*)d_in[i]; };

    int NJ = in_sizes[0] / 16;
    int NS = in_sizes[1] / 8;
    int NM = in_sizes[2] / 8;
    int NR = in_sizes[3] / 8;
    int NA = in_sizes[14] / 3;
    int NsMax = NS > NM ? NS : NM; if (NR > NsMax) NsMax = NR;
    int Emax = 0; for (int i = 0; i < 10; ++i) if (nE[i] > Emax) Emax = nE[i];

    // workspace carve (all chunks are multiples of 4 floats -> float4 aligned)
    float* wsp = (float*)d_ws;
    auto alloc = [&](size_t nf) -> float* { float* p = wsp; wsp += nf; return p; };
    float* hj    = alloc((size_t)NJ * 128);
    float* big2  = alloc((size_t)NJ * 128);   // out_j (job blocks) / P_big (other blocks)
    float* hs    = alloc((size_t)NS * 128);
    float* hm    = alloc((size_t)NM * 128);
    float* hr    = alloc((size_t)NR * 128);
    float* out_s = alloc((size_t)NS * 128);
    float* out_m = alloc((size_t)NM * 128);
    float* out_r = alloc((size_t)NR * 128);
    float* Psml  = alloc((size_t)NsMax * 128);
    float* a_sb  = alloc((size_t)NJ * 4);
    float* a_db  = alloc((size_t)NJ * 4);
    float* mxb   = alloc((size_t)NJ * 4);
    float* denb  = alloc((size_t)NJ * 4);
    float* ebuf  = alloc((size_t)Emax * 4);
    float* Vbuf  = alloc(512);
    float* csum  = alloc(128);
    float* hglob = alloc(128);

    auto cdiv = [](long long a, long long b) -> unsigned int { return (unsigned int)((a + b - 1) / b); };

    // one GAT conv: adds messages into `outb` (which holds residual + biases)
    auto run_gat = [&](const float* xs, int Ns, const float* xd, int Nd,
                       const int* edge, int E, int pb /*att_dst,att_src,bias,w*/,
                       float* Pbuf, float* outb)
    {
        const float* att_dst = PF(pb + 0);
        const float* att_src = PF(pb + 1);
        const float* Wt      = PF(pb + 3);
        k_gemm128_wmma<<<dim3(cdiv(Ns, 64)), dim3(32, 8), 0, stream>>>(xs, Wt, nullptr, Pbuf, Ns, 128, 0);
        k_head_reduce<<<cdiv(Ns, 256), 256, 0, stream>>>(Pbuf, att_src, a_sb, Ns);
        k_fold_att<<<2, 256, 0, stream>>>(Wt, att_dst, Vbuf);
        k_rowdot4<<<cdiv(Nd, 256), 256, 0, stream>>>(xd, Vbuf, a_db, Nd);
        k_fill<<<256, 256, 0, stream>>>(mxb, -3.0e38f, (long long)Nd * 4);
        k_fill<<<256, 256, 0, stream>>>(denb, 0.0f, (long long)Nd * 4);
        k_edge1<<<cdiv((long long)E * 4, 256), 256, 0, stream>>>(edge, E, a_sb, a_db, ebuf, mxb);
        k_edge2<<<cdiv((long long)E * 4, 256), 256, 0, stream>>>(edge, E, ebuf, mxb, denb);
        k_edge3<<<cdiv((long long)E * 32, 256), 256, 0, stream>>>(edge, E, ebuf, denb, Pbuf, outb);
    };

    // ---- input projections: relu(x @ W + b) ----
    k_gemm128_wmma<<<dim3(cdiv(NJ, 64)), dim3(32, 8), 0, stream>>>(x_job, PF(63), PF(62), hj, NJ, 16, 1);
    k_gemm128_wmma<<<dim3(cdiv(NS, 64)), dim3(32, 8), 0, stream>>>(x_sta, PF(69), PF(68), hs, NS, 8, 1);
    k_gemm128_wmma<<<dim3(cdiv(NM, 64)), dim3(32, 8), 0, stream>>>(x_mac, PF(65), PF(64), hm, NM, 8, 1);
    k_gemm128_wmma<<<dim3(cdiv(NR, 64)), dim3(32, 8), 0, stream>>>(x_rob, PF(67), PF(66), hr, NR, 8, 1);

    // param base indices (sorted-key jax flattening; each GAT = att_dst,att_src,bias,w)
    // job1: can_load=18 execute=22 hold=26 loaded=30 norm.b=34 norm.g=35 will_execute=36
    // job2: can_load=40 execute=44 hold=48 loaded=52 norm.b=56 norm.g=57 will_execute=58
    auto job_block = [&](int cl, int ex, int ho, int ld, int nb, int we)
    {
        k_init_out<<<cdiv((long long)NJ * 32, 256), 256, 0, stream>>>(
            (float4*)big2, (const float4*)hj, PF(cl + 2), PF(ld + 2), PF(we + 2),
            PF(ex + 2), PF(ho + 2), (long long)NJ * 32);
        run_gat(hs, NS, hj, NJ, eg[0], nE[0], cl, Psml, big2);  // can_load
        run_gat(hs, NS, hj, NJ, eg[1], nE[1], ld, Psml, big2);  // loaded
        run_gat(hm, NM, hj, NJ, eg[2], nE[2], we, Psml, big2);  // will_execute
        run_gat(hm, NM, hj, NJ, eg[3], nE[3], ex, Psml, big2);  // execute
        run_gat(hr, NR, hj, NJ, eg[4], nE[4], ho, Psml, big2);  // hold
        k_ln_relu<<<NJ, 128, 0, stream>>>(big2, PF(nb + 1), PF(nb + 0), hj, NJ);
    };

    // other1: cbl=70 exby=74 holdby=78 ldin=82 needs=86 norm_m.b=90 norm_r.b=92 norm_s.b=94
    // other2: cbl=96 exby=100 holdby=104 ldin=108 needs=112 norm_m.b=116 norm_r.b=118 norm_s.b=120
    auto other_block = [&](int cbl, int exby, int hob, int ldin, int nds,
                           int nmb, int nrb, int nsb)
    {
        k_init_out<<<cdiv((long long)NS * 32, 256), 256, 0, stream>>>(
            (float4*)out_s, (const float4*)hs, PF(cbl + 2), PF(ldin + 2),
            nullptr, nullptr, nullptr, (long long)NS * 32);
        k_init_out<<<cdiv((long long)NM * 32, 256), 256, 0, stream>>>(
            (float4*)out_m, (const float4*)hm, PF(nds + 2), PF(exby + 2),
            nullptr, nullptr, nullptr, (long long)NM * 32);
        k_init_out<<<cdiv((long long)NR * 32, 256), 256, 0, stream>>>(
            (float4*)out_r, (const float4*)hr, PF(hob + 2), nullptr,
            nullptr, nullptr, nullptr, (long long)NR * 32);
        run_gat(hj, NJ, hs, NS, eg[5], nE[5], cbl,  big2, out_s);  // could_be_loaded
        run_gat(hj, NJ, hs, NS, eg[6], nE[6], ldin, big2, out_s);  // loaded_in
        run_gat(hj, NJ, hm, NM, eg[7], nE[7], nds,  big2, out_m);  // needs
        run_gat(hj, NJ, hm, NM, eg[8], nE[8], exby, big2, out_m);  // executed_by
        run_gat(hj, NJ, hr, NR, eg[9], nE[9], hob,  big2, out_r);  // hold_by
        k_ln_relu<<<NS, 128, 0, stream>>>(out_s, PF(nsb + 1), PF(nsb + 0), hs, NS);
        k_ln_relu<<<NM, 128, 0, stream>>>(out_m, PF(nmb + 1), PF(nmb + 0), hm, NM);
        k_ln_relu<<<NR, 128, 0, stream>>>(out_r, PF(nrb + 1), PF(nrb + 0), hr, NR);
    };

    job_block(18, 22, 26, 30, 34, 36);
    other_block(70, 74, 78, 82, 86, 90, 92, 94);
    job_block(40, 44, 48, 52, 56, 58);
    other_block(96, 100, 104, 108, 112, 116, 118, 120);

    // ---- global head ----
    k_fill<<<1, 128, 0, stream>>>(csum, 0.0f, 128);
    k_colsum<<<256, 128, 0, stream>>>(hj, csum, NJ);
    k_global_head<<<1, 128, 0, stream>>>(hs, hm, hr, csum, 1.0f / (float)NJ,
                                         PF(17), PF(16), hglob);   // global.w, global.b

    // ---- Q head ----
    k_qhead<<<NA, 64, 0, stream>>>(actions, alpha, hj, hglob,
                                   PF(123), PF(122),   // q1.w, q1.b
                                   PF(125), PF(124),   // q2.w, q2.b
                                   PF(127), PF(126),   // q3.w, q3.b
                                   (float*)d_out, NA);
}